// SocialLSTM_27513560498699
// MI455X (gfx1250) — compile-verified
//
#include <hip/hip_runtime.h>
#include <math.h>

typedef __attribute__((ext_vector_type(2))) float v2f;
typedef __attribute__((ext_vector_type(8))) float v8f;

#define OBS_T 8
#define PRED_T 12
#define NSTEP 20
#define EDIM 64
#define HDIM 128
#define PP 32
#define NPED 1024
#define DIN 256   // E + H + E

static __device__ __forceinline__ v8f wmma_f32(v8f c, float ax, float ay, float bx, float by) {
  v2f a; a[0] = ax; a[1] = ay;
  v2f b; b[0] = bx; b[1] = by;
  return __builtin_amdgcn_wmma_f32_16x16x4_f32(false, a, false, b, (short)0, c, false, false);
}

__global__ __launch_bounds__(256, 1)
void social_lstm_kernel(const float* __restrict__ obs_rel,   // [8,1024,2]
                        const float* __restrict__ obs,       // [8,1024,2]
                        const int*   __restrict__ sse,       // [32,2]
                        const float* __restrict__ We,        // [2,64]
                        const float* __restrict__ be,        // [64]
                        const float* __restrict__ Wp,        // [8192,64] rows = bin*128 + h
                        const float* __restrict__ bp,        // [64]
                        const float* __restrict__ Wi, const float* __restrict__ bi,
                        const float* __restrict__ Wf, const float* __restrict__ bf,
                        const float* __restrict__ Wo, const float* __restrict__ bo,
                        const float* __restrict__ Wg, const float* __restrict__ bg,
                        const float* __restrict__ Wout,      // [128,2]
                        const float* __restrict__ bout,      // [2]
                        float* __restrict__ out)             // [12,1024,2]
{
  extern __shared__ float smem[];
  float* comb  = smem;                    // [32][256] : 0:64 emb | 64:192 h | 192:256 s
  float* cst   = comb + PP*DIN;           // [32][128] cell state
  float* gbuf  = cst + PP*HDIM;           // 16384 floats: union {Cchunk[8][32][64], gates[4][32][128]}
  int*   binsS = (int*)(gbuf + 16384);    // [32][32] pair bins (-1 invalid)
  float* posS  = (float*)(binsS + PP*PP); // [32][2]
  float* pvS   = posS + 2*PP;             // [32][2]

  const int tid   = (int)threadIdx.x;
  const int wv    = tid >> 5;
  const int lane  = tid & 31;
  const int mrow  = lane & 15;   // A-frag: M row within tile
  const int khalf = lane >> 4;   // A/B-frag: K pair select
  const int nloc  = lane & 15;   // B/C-frag: N col within tile
  const int p0    = sse[blockIdx.x * 2];

  for (int r = tid; r < PP*DIN;  r += 256) comb[r] = 0.f;
  for (int r = tid; r < PP*HDIM; r += 256) cst[r]  = 0.f;
  __syncthreads();

  const float* Wgate[4] = {Wi, Wf, Wo, Wg};
  const float gnorm = 2.0f / 7.0f;   // NEIGH/(G-1)

  for (int step = 0; step < NSTEP; ++step) {
    if (step < OBS_T) {
      // curr_pos = obs[t]; emb = obs_rel[t] @ We + be
      if (tid < 2*PP) posS[tid] = obs[(size_t)step*(NPED*2) + p0*2 + tid];
      {
        int p = tid >> 3, es = (tid & 7) * 8;
        float x0 = obs_rel[(size_t)step*(NPED*2) + (size_t)(p0+p)*2 + 0];
        float x1 = obs_rel[(size_t)step*(NPED*2) + (size_t)(p0+p)*2 + 1];
        for (int r = 0; r < 8; ++r) {
          int e = es + r;
          comb[p*DIN + e] = x0*We[e] + x1*We[EDIM + e] + be[e];
        }
      }
      __syncthreads();
      // pair bins from new pos
      for (int q = 0; q < 4; ++q) {
        int pr = tid*4 + q;
        int i = pr >> 5, j = pr & 31;
        float rx = posS[j*2+0] - posS[i*2+0];
        float ry = posS[j*2+1] - posS[i*2+1];
        float cx = fminf(fmaxf(rx / gnorm, -4.f), 4.f) + 4.f;
        float cy = fminf(fmaxf(ry / gnorm, -4.f), 4.f) + 4.f;
        int gx = (int)cx, gy = (int)cy;
        binsS[pr] = (gx < 8 && gy < 8 && i != j) ? (gy*8 + gx) : -1;
      }
      __syncthreads();
    } else {
      // bins use pos from BEFORE this step's update
      for (int q = 0; q < 4; ++q) {
        int pr = tid*4 + q;
        int i = pr >> 5, j = pr & 31;
        float rx = posS[j*2+0] - posS[i*2+0];
        float ry = posS[j*2+1] - posS[i*2+1];
        float cx = fminf(fmaxf(rx / gnorm, -4.f), 4.f) + 4.f;
        float cy = fminf(fmaxf(ry / gnorm, -4.f), 4.f) + 4.f;
        int gx = (int)cx, gy = (int)cy;
        binsS[pr] = (gx < 8 && gy < 8 && i != j) ? (gy*8 + gx) : -1;
      }
      __syncthreads();
      // pv = h @ Wout + bout; emit pred; pos += pv
      if (tid < 2*PP) {
        int p = tid >> 1, d = tid & 1;
        float acc = bout[d];
        for (int k = 0; k < HDIM; ++k) acc += comb[p*DIN + EDIM + k] * Wout[k*2 + d];
        pvS[tid] = acc;
        out[((size_t)(step - OBS_T)*NPED + (size_t)(p0 + p))*2 + d] = acc;
        posS[tid] += acc;
      }
      __syncthreads();
      // emb = pv @ We + be
      {
        int p = tid >> 3, es = (tid & 7) * 8;
        float x0 = pvS[p*2+0], x1 = pvS[p*2+1];
        for (int r = 0; r < 8; ++r) {
          int e = es + r;
          comb[p*DIN + e] = x0*We[e] + x1*We[EDIM + e] + be[e];
        }
      }
      __syncthreads();
    }

    // ---- social pooling: pool[i] = sum_j (h[j] @ Wp_bin(i,j)) ----
    float sacc[8];
    for (int r = 0; r < 8; ++r) sacc[r] = 0.f;

    for (int cb = 0; cb < 8; ++cb) {            // 8 bins per chunk
      // GEMM: h[32,128] @ Wp[cb*8..cb*8+8][128,64]  -> gbuf as C[8][32][64]
      for (int q = 0; q < 8; ++q) {
        int tt = wv*8 + q;
        int mt = tt & 1, nt = tt >> 1;          // N = 512 -> nt 0..31
        int n  = nt*16 + nloc;
        int bl = n >> 6;                        // bin within chunk
        int e  = n & 63;
        const float* B = Wp + ((size_t)(cb*8 + bl) * HDIM) * EDIM + e;
        const float* A = comb + (mt*16 + mrow)*DIN + EDIM;   // h slice of comb
        v8f acc = {};
        for (int k = 0; k < 32; ++k) {
          int kk = k*4 + khalf*2;
          acc = wmma_f32(acc, A[kk], A[kk+1],
                         B[(size_t)kk * EDIM], B[(size_t)(kk+1) * EDIM]);
        }
        float* C = gbuf + bl*(PP*EDIM) + (mt*16 + khalf*8)*EDIM + e;
        for (int r = 0; r < 8; ++r) C[r*EDIM] = acc[r];
      }
      __syncthreads();
      // gather-sum rows of C whose pair-bin falls in this chunk
      {
        int i = tid >> 3, es = (tid & 7) * 8;
        for (int j = 0; j < PP; ++j) {
          int bn = binsS[i*PP + j] - cb*8;
          if ((unsigned)bn < 8u) {
            const float* src = gbuf + bn*(PP*EDIM) + j*EDIM + es;
            for (int r = 0; r < 8; ++r) sacc[r] += src[r];
          }
        }
      }
      __syncthreads();
    }
    // s = relu(pool + bp) into comb[:,192:256]
    {
      int i = tid >> 3, es = (tid & 7) * 8;
      for (int r = 0; r < 8; ++r) {
        float v = sacc[r] + bp[es + r];
        comb[i*DIN + (EDIM + HDIM) + es + r] = v > 0.f ? v : 0.f;
      }
    }
    __syncthreads();

    // ---- gate GEMMs: comb[32,256] @ W[256,128] for i,f,o,g ----
    for (int q = 0; q < 8; ++q) {
      int tt  = wv*8 + q;                       // 64 tiles: 4 gates x (2x8)
      int g   = tt >> 4;
      int rem = tt & 15;
      int mt  = rem & 1, nt = rem >> 1;
      int n   = nt*16 + nloc;
      const float* W = Wgate[g];
      const float* A = comb + (mt*16 + mrow)*DIN;
      v8f acc = {};
      for (int k = 0; k < 64; ++k) {
        int kk = k*4 + khalf*2;
        acc = wmma_f32(acc, A[kk], A[kk+1],
                       W[(size_t)kk * HDIM + n], W[(size_t)(kk+1) * HDIM + n]);
      }
      float* C = gbuf + g*(PP*HDIM) + (mt*16 + khalf*8)*HDIM + n;
      for (int r = 0; r < 8; ++r) C[r*HDIM] = acc[r];
    }
    __syncthreads();

    // ---- elementwise LSTM cell ----
    {
      int p = tid >> 3, ns = (tid & 7) * 16;
      for (int r = 0; r < 16; ++r) {
        int n = ns + r;
        float pi = gbuf[0*(PP*HDIM) + p*HDIM + n] + bi[n];
        float pf = gbuf[1*(PP*HDIM) + p*HDIM + n] + bf[n];
        float po = gbuf[2*(PP*HDIM) + p*HDIM + n] + bo[n];
        float pg = gbuf[3*(PP*HDIM) + p*HDIM + n] + bg[n];
        float ig = 1.f / (1.f + expf(-pi));
        float fg = 1.f / (1.f + expf(-pf));
        float og = 1.f / (1.f + expf(-po));
        float cc = fg * cst[p*HDIM + n] + ig * tanhf(pg);
        cst[p*HDIM + n] = cc;
        comb[p*DIN + EDIM + n] = og * tanhf(cc);
      }
    }
    __syncthreads();
  }
}

extern "C" void kernel_launch(void* const* d_in, const int* in_sizes, int n_in,
                              void* d_out, int out_size, void* d_ws, size_t ws_size,
                              hipStream_t stream) {
  (void)in_sizes; (void)n_in; (void)out_size; (void)d_ws; (void)ws_size;
  const float* obs_rel = (const float*)d_in[0];
  const float* obs     = (const float*)d_in[1];
  const int*   sse     = (const int*)  d_in[2];
  const float* We   = (const float*)d_in[3];
  const float* be   = (const float*)d_in[4];
  const float* Wp   = (const float*)d_in[5];
  const float* bp   = (const float*)d_in[6];
  const float* Wi   = (const float*)d_in[7];
  const float* bi   = (const float*)d_in[8];
  const float* Wf   = (const float*)d_in[9];
  const float* bf   = (const float*)d_in[10];
  const float* Wo   = (const float*)d_in[11];
  const float* bo   = (const float*)d_in[12];
  const float* Wg   = (const float*)d_in[13];
  const float* bg   = (const float*)d_in[14];
  const float* Wout = (const float*)d_in[15];
  const float* bout = (const float*)d_in[16];
  float* out = (float*)d_out;

  // LDS: comb 32*256 + c 32*128 + union 16384 floats + bins 1024 ints + pos/pv 128 floats
  const size_t shmem = (size_t)(PP*DIN + PP*HDIM + 16384 + 2*PP + 2*PP) * sizeof(float)
                     + (size_t)(PP*PP) * sizeof(int);
  hipFuncSetAttribute((const void*)social_lstm_kernel,
                      hipFuncAttributeMaxDynamicSharedMemorySize, (int)shmem);
  social_lstm_kernel<<<32, 256, shmem, stream>>>(
      obs_rel, obs, sse, We, be, Wp, bp,
      Wi, bi, Wf, bf, Wo, bo, Wg, bg, Wout, bout, out);
}